// HopGatedGATv2Conv_24146306138834
// MI455X (gfx1250) — compile-verified
//
#include <hip/hip_runtime.h>
#include <math.h>

typedef __attribute__((ext_vector_type(2))) float v2f;
typedef __attribute__((ext_vector_type(8))) float v8f;

#define NEG_SLOPE 0.2f

// Monotonic float<->uint encoding so atomicMax(uint) == float max.
__device__ __forceinline__ unsigned fenc(float f) {
  unsigned u = __float_as_uint(f);
  return (u & 0x80000000u) ? ~u : (u | 0x80000000u);
}
__device__ __forceinline__ float fdec(unsigned u) {
  return __uint_as_float((u & 0x80000000u) ? (u ^ 0x80000000u) : ~u);
}

// out[N,128] = x[N,128] @ w[128,128] + bias   (fp32 WMMA 16x16x4, wave per 16x16 tile)
// blockIdx.y selects (w_l,b_l,xl) vs (w_r,b_r,xr).
__global__ void gemm128_wmma(const float* __restrict__ x,
                             const float* __restrict__ wl, const float* __restrict__ bl,
                             const float* __restrict__ wr, const float* __restrict__ br,
                             float* __restrict__ xl, float* __restrict__ xr,
                             int N, int rowTiles) {
  const int wave = threadIdx.x >> 5;
  const int lane = threadIdx.x & 31;
  const int tileId = blockIdx.x * 8 + wave;
  const int tilesPerMat = rowTiles * 8;           // 128/16 = 8 col tiles
  if (tileId >= tilesPerMat) return;
  const float* w  = blockIdx.y ? wr : wl;
  const float* bs = blockIdx.y ? br : bl;
  float* outp     = blockIdx.y ? xr : xl;
  const int tm  = (tileId >> 3) * 16;
  const int tn  = (tileId & 7) * 16;
  const int hi2 = (lane >> 4) * 2;                // lane-half -> K offset {0,2}
  const int l16 = lane & 15;
  int m = tm + l16; if (m >= N) m = N - 1;        // clamp (stores are guarded)
  const int n = tn + l16;
  v8f acc = {};
  for (int k0 = 0; k0 < 128; k0 += 4) {
    const int ka = k0 + hi2;
    v2f a, b;
    // A 16x4 (ISA layout): VGPR0/1 = K {ka, ka+1}, lane-half adds +2
    a.x = x[(size_t)m * 128 + ka];
    a.y = x[(size_t)m * 128 + ka + 1];
    // B 4x16: row (K) striped across lanes per VGPR, lane-half adds +2
    b.x = w[(size_t)ka * 128 + n];
    b.y = w[(size_t)(ka + 1) * 128 + n];
    acc = __builtin_amdgcn_wmma_f32_16x16x4_f32(false, a, false, b, (short)0, acc,
                                                false, false);
  }
  const float bias = bs[n];
#pragma unroll
  for (int r = 0; r < 8; ++r) {                   // C/D: VGPR r -> M=r / M=r+8
    const int row = tm + r + (lane >> 4) * 8;
    if (row < N) outp[(size_t)row * 128 + n] = acc[r] + bias;
  }
}

// Gate: weights[n,:] = softmax(x[n]@gate_w + gate_b), wave per node.
__global__ void gate_softmax(const float* __restrict__ x, const float* __restrict__ gw,
                             const float* __restrict__ gb, float* __restrict__ wgt, int N) {
  const int lane = threadIdx.x & 31;
  const int n = blockIdx.x * 8 + (threadIdx.x >> 5);
  if (n >= N) return;
  float s0 = 0.f, s1 = 0.f, s2 = 0.f;
  for (int i = lane; i < 128; i += 32) {
    const float xv = x[(size_t)n * 128 + i];
    s0 += xv * gw[i * 3 + 0];
    s1 += xv * gw[i * 3 + 1];
    s2 += xv * gw[i * 3 + 2];
  }
  for (int off = 16; off; off >>= 1) {
    s0 += __shfl_xor(s0, off, 32);
    s1 += __shfl_xor(s1, off, 32);
    s2 += __shfl_xor(s2, off, 32);
  }
  if (lane == 0) {
    s0 += gb[0]; s1 += gb[1]; s2 += gb[2];
    const float mx = fmaxf(s0, fmaxf(s1, s2));
    const float e0 = __expf(s0 - mx), e1 = __expf(s1 - mx), e2 = __expf(s2 - mx);
    const float inv = 1.0f / (e0 + e1 + e2);
    wgt[n * 3 + 0] = e0 * inv;
    wgt[n * 3 + 1] = e1 * inv;
    wgt[n * 3 + 2] = e2 * inv;
  }
}

// Pass A: logit[e,h] = sum_c leakyrelu(xl[src,h,c]+xr[dst,h,c]) * att[h,c]
//         + atomic segment-max into menc[dst,h]. Wave per edge, lane = channel.
__global__ void edge_logit_max(const float* __restrict__ xl, const float* __restrict__ xr,
                               const int* __restrict__ srcA, const int* __restrict__ dstA,
                               const float* __restrict__ att, float* __restrict__ elog,
                               unsigned* __restrict__ menc, int Ereal, int nE) {
  const int lane = threadIdx.x & 31;
  const int e = blockIdx.x * 8 + (threadIdx.x >> 5);
  if (e >= nE) return;
  int src, dst;
  if (e < Ereal) { src = srcA[e]; dst = dstA[e]; } else { src = e - Ereal; dst = src; }
  const float* pl = xl + (size_t)src * 128;
  const float* pr = xr + (size_t)dst * 128;
  float g, a0, a1, a2, a3;
  g = pl[lane]      + pr[lane];      g = g > 0.f ? g : NEG_SLOPE * g; a0 = g * att[lane];
  g = pl[32 + lane] + pr[32 + lane]; g = g > 0.f ? g : NEG_SLOPE * g; a1 = g * att[32 + lane];
  g = pl[64 + lane] + pr[64 + lane]; g = g > 0.f ? g : NEG_SLOPE * g; a2 = g * att[64 + lane];
  g = pl[96 + lane] + pr[96 + lane]; g = g > 0.f ? g : NEG_SLOPE * g; a3 = g * att[96 + lane];
  for (int off = 16; off; off >>= 1) {
    a0 += __shfl_xor(a0, off, 32);
    a1 += __shfl_xor(a1, off, 32);
    a2 += __shfl_xor(a2, off, 32);
    a3 += __shfl_xor(a3, off, 32);
  }
  if (lane == 0) {
    elog[(size_t)e * 4 + 0] = a0; elog[(size_t)e * 4 + 1] = a1;
    elog[(size_t)e * 4 + 2] = a2; elog[(size_t)e * 4 + 3] = a3;
    atomicMax(&menc[dst * 4 + 0], fenc(a0));
    atomicMax(&menc[dst * 4 + 1], fenc(a1));
    atomicMax(&menc[dst * 4 + 2], fenc(a2));
    atomicMax(&menc[dst * 4 + 3], fenc(a3));
  }
}

// Pass B: a = exp(logit - m[dst]) (in-place over elog); denom[dst,h] += a.
__global__ void edge_exp_denom(const int* __restrict__ dstA,
                               float* __restrict__ elog, const unsigned* __restrict__ menc,
                               float* __restrict__ denom, int Ereal, int nE) {
  const long long idx = (long long)blockIdx.x * blockDim.x + threadIdx.x;
  if (idx >= (long long)nE * 4) return;
  const int e = (int)(idx >> 2), h = (int)(idx & 3);
  const int dst = (e < Ereal) ? dstA[e] : (e - Ereal);
  const float a = __expf(elog[idx] - fdec(menc[dst * 4 + h]));
  elog[idx] = a;
  atomicAdd(&denom[dst * 4 + h], a);
}

// Pass C: msg[dst,c] += (1/H) * sum_h alpha[e,h]*xl[src,h,c]. Wave per edge.
__global__ void edge_aggregate(const float* __restrict__ xl,
                               const int* __restrict__ srcA, const int* __restrict__ dstA,
                               const float* __restrict__ ea, const float* __restrict__ denom,
                               float* __restrict__ msg, int Ereal, int nE) {
  const int lane = threadIdx.x & 31;
  const int e = blockIdx.x * 8 + (threadIdx.x >> 5);
  if (e >= nE) return;
  int src, dst;
  if (e < Ereal) { src = srcA[e]; dst = dstA[e]; } else { src = e - Ereal; dst = src; }
  const float al0 = ea[(size_t)e * 4 + 0] / denom[dst * 4 + 0];
  const float al1 = ea[(size_t)e * 4 + 1] / denom[dst * 4 + 1];
  const float al2 = ea[(size_t)e * 4 + 2] / denom[dst * 4 + 2];
  const float al3 = ea[(size_t)e * 4 + 3] / denom[dst * 4 + 3];
  const float* pl = xl + (size_t)src * 128;
  const float s = al0 * pl[lane] + al1 * pl[32 + lane] +
                  al2 * pl[64 + lane] + al3 * pl[96 + lane];
  atomicAdd(&msg[(size_t)dst * 32 + lane], 0.25f * s);
}

// out[n,c] = sum_k wgt[n,k] * (msg_k[n,c] + conv_bias[k,c])
__global__ void combine_out(const float* __restrict__ msg0, const float* __restrict__ msg1,
                            const float* __restrict__ msg2, const float* __restrict__ cb,
                            const float* __restrict__ wgt, float* __restrict__ out, int N) {
  const int idx = blockIdx.x * blockDim.x + threadIdx.x;
  if (idx >= N * 32) return;
  const int n = idx >> 5, c = idx & 31;
  const float w0 = wgt[n * 3 + 0], w1 = wgt[n * 3 + 1], w2 = wgt[n * 3 + 2];
  out[idx] = w0 * (msg0[idx] + cb[c]) +
             w1 * (msg1[idx] + cb[32 + c]) +
             w2 * (msg2[idx] + cb[64 + c]);
}

extern "C" void kernel_launch(void* const* d_in, const int* in_sizes, int n_in,
                              void* d_out, int out_size, void* d_ws, size_t ws_size,
                              hipStream_t stream) {
  const float* x = (const float*)d_in[0];
  const int N = in_sizes[0] / 128;
  const int* ei[3] = {(const int*)d_in[1], (const int*)d_in[2], (const int*)d_in[3]};
  int E[3];
  for (int k = 0; k < 3; ++k) E[k] = in_sizes[1 + k] / 2;   // [2,E] stacked src,dst
  const float* wl  = (const float*)d_in[4];
  const float* bl  = (const float*)d_in[5];
  const float* wr  = (const float*)d_in[6];
  const float* br  = (const float*)d_in[7];
  const float* att = (const float*)d_in[8];    // [3,4,32]
  const float* cb  = (const float*)d_in[9];    // [3,32]
  const float* gw  = (const float*)d_in[10];   // [128,3]
  const float* gb  = (const float*)d_in[11];   // [3]

  int maxE = 0;
  for (int k = 0; k < 3; ++k) {
    const int nE = E[k] + (k == 0 ? N : 0);
    if (nE > maxE) maxE = nE;
  }

  // Workspace carve (~86 MB for the reference sizes)
  char* base = (char*)d_ws;
  size_t off = 0;
  auto carve = [&](size_t bytes) -> char* {
    char* p = base + off;
    off = (off + bytes + 255) & ~(size_t)255;
    return p;
  };
  float*    xl   = (float*)carve((size_t)N * 128 * 4);
  float*    xr   = (float*)carve((size_t)N * 128 * 4);
  float*    wgt  = (float*)carve((size_t)N * 3 * 4);
  unsigned* menc = (unsigned*)carve((size_t)N * 4 * 4);
  float*    den  = (float*)carve((size_t)N * 4 * 4);
  float*    msg  = (float*)carve((size_t)N * 32 * 4 * 3);
  float*    elog = (float*)carve((size_t)maxE * 4 * 4);
  (void)ws_size; (void)n_in; (void)out_size;

  // 1) Projections x_l, x_r via fp32 WMMA
  {
    const int rowTiles = (N + 15) / 16;
    dim3 grid((unsigned)rowTiles, 2);              // 8 waves/block = 8 tiles/block
    gemm128_wmma<<<grid, 256, 0, stream>>>(x, wl, bl, wr, br, xl, xr, N, rowTiles);
  }
  // 2) Gate softmax weights
  gate_softmax<<<(N + 7) / 8, 256, 0, stream>>>(x, gw, gb, wgt, N);
  // 3) Zero message accumulators
  hipMemsetAsync(msg, 0, (size_t)N * 32 * 4 * 3, stream);
  // 4) Hops
  for (int k = 0; k < 3; ++k) {
    const int Ereal = E[k];
    const int nE = Ereal + (k == 0 ? N : 0);       // hop 0 adds self-loops
    hipMemsetAsync(menc, 0, (size_t)N * 4 * 4, stream);
    hipMemsetAsync(den, 0, (size_t)N * 4 * 4, stream);
    const int* srcA = ei[k];
    const int* dstA = ei[k] + Ereal;
    edge_logit_max<<<(nE + 7) / 8, 256, 0, stream>>>(xl, xr, srcA, dstA,
                                                     att + (size_t)k * 128, elog, menc,
                                                     Ereal, nE);
    const long long tot = (long long)nE * 4;
    edge_exp_denom<<<(unsigned)((tot + 255) / 256), 256, 0, stream>>>(dstA, elog, menc,
                                                                      den, Ereal, nE);
    edge_aggregate<<<(nE + 7) / 8, 256, 0, stream>>>(xl, srcA, dstA, elog, den,
                                                     msg + (size_t)k * N * 32, Ereal, nE);
  }
  // 5) Gated combine
  combine_out<<<(N * 32 + 255) / 256, 256, 0, stream>>>(msg, msg + (size_t)N * 32,
                                                        msg + (size_t)2 * N * 32, cb, wgt,
                                                        (float*)d_out, N);
}